// ConvolutionalControl_52338471469497
// MI455X (gfx1250) — compile-verified
//
#include <hip/hip_runtime.h>

typedef __attribute__((ext_vector_type(2))) float v2f;
typedef __attribute__((ext_vector_type(8))) float v8f;

// ---------------------------------------------------------------------------
// Hardware float atomic add (non-returning -> global_atomic_add_f32, STOREcnt)
// ---------------------------------------------------------------------------
__device__ __forceinline__ void atomic_add_f32(float* p, float v) {
#if defined(__HIP_DEVICE_COMPILE__)
    unsafeAtomicAdd(p, v);
#else
    atomicAdd(p, v);
#endif
}

// ---------------------------------------------------------------------------
// Zero-fill (harness poisons d_out / d_ws with 0xAA)
// ---------------------------------------------------------------------------
__global__ void zero_kernel(float* __restrict__ p, size_t n) {
    size_t i = (size_t)blockIdx.x * blockDim.x + threadIdx.x;
    if (i < n) p[i] = 0.0f;
}

// ---------------------------------------------------------------------------
// deg[v] = number of incoming edges (float; exact for counts << 2^24)
// ---------------------------------------------------------------------------
__global__ void degree_kernel(const long long* __restrict__ dst,
                              float* __restrict__ deg, int E) {
    int e = blockIdx.x * blockDim.x + threadIdx.x;
    if (e < E) atomic_add_f32(&deg[dst[e]], 1.0f);
}

// ---------------------------------------------------------------------------
// h = x @ W^T + b  via V_WMMA_F32_16X16X4_F32 (full fp32 precision).
// One wave32 computes a 16-row x 128-col strip: 8 column tiles, K stepped by 4.
//
// A-tile (16x4 f32): lanes 0-15 hold rows M=0..15 with VGPR0=K, VGPR1=K+1;
//                    lanes 16-31 hold the same rows with K+2, K+3.
// B-tile (4x16 f32): B[k][n] = W[col0+n][k]  (mirrored layout) -> identical
//                    contiguous-pair load pattern from rows of W.
// C/D (16x16 f32):   VGPR v, lanes 0-15 -> M=v; lanes 16-31 -> M=v+8.
// ---------------------------------------------------------------------------
__global__ __launch_bounds__(256) void gemm_wmma_kernel(
    const float* __restrict__ x, const float* __restrict__ W,
    const float* __restrict__ bias, float* __restrict__ h, int N) {
    const int lane  = threadIdx.x & 31;
    const int wave  = threadIdx.x >> 5;
    const int strip = blockIdx.x * 8 + wave;        // one 16-row strip per wave
    const int row0  = strip * 16;
    if (row0 >= N) return;

    const int m    = lane & 15;                     // row (A) / col (B) within tile
    const int koff = (lane >> 4) * 2;               // K sub-offset from lane half

    // Clamp for a possible ragged tail strip (loads only; stores predicated).
    const int rowA = (row0 + m < N) ? (row0 + m) : (N - 1);
    const float* __restrict__ xrow = x + (size_t)rowA * 128 + koff;

    v8f acc[8];
#pragma unroll
    for (int t = 0; t < 8; ++t) acc[t] = (v8f)0.0f;

#pragma unroll 4
    for (int k = 0; k < 128; k += 4) {
        v2f a = *(const v2f*)(xrow + k);
#pragma unroll
        for (int t = 0; t < 8; ++t) {
            // B[k'][n] = W[t*16 + n][k'],  n = lane&15, k' = k+koff(+1)
            v2f bv = *(const v2f*)(W + (size_t)(t * 16 + m) * 128 + k + koff);
            acc[t] = __builtin_amdgcn_wmma_f32_16x16x4_f32(
                false, a, false, bv, (short)0, acc[t], false, false);
        }
    }

    // Store D with fused bias. For a fixed VGPR v, lanes 0-15 write one
    // contiguous 64B run (row rbase+v), lanes 16-31 another (row rbase+8+v).
    const int rbase = row0 + ((lane >> 4) << 3);
#pragma unroll
    for (int t = 0; t < 8; ++t) {
        const int col = t * 16 + m;
        const float bb = bias[col];
#pragma unroll
        for (int v = 0; v < 8; ++v) {
            const int r = rbase + v;
            if (r < N) h[(size_t)r * 128 + col] = acc[t][v] + bb;
        }
    }
}

// ---------------------------------------------------------------------------
// out[dst] += (1/deg[dst]) * h[src]   -- 8 edges/block, 32 lanes/edge,
// float4 gather from L2-resident h, 4 hardware float atomics per lane.
// ---------------------------------------------------------------------------
__global__ __launch_bounds__(256) void scatter_kernel(
    const long long* __restrict__ src, const long long* __restrict__ dst,
    const float* __restrict__ h, const float* __restrict__ deg,
    float* __restrict__ out, int E) {
    const int lane = threadIdx.x & 31;
    const int sub  = threadIdx.x >> 5;
    const long long e = (long long)blockIdx.x * 8 + sub;
    if (e >= E) return;

    const long long s = src[e];
    const long long d = dst[e];
    const float w = 1.0f / deg[d];          // deg[d] >= 1: this edge targets d

    const float4 hv = *(const float4*)(h + s * 128 + lane * 4);
    float* op = out + d * 128 + lane * 4;
    atomic_add_f32(op + 0, w * hv.x);
    atomic_add_f32(op + 1, w * hv.y);
    atomic_add_f32(op + 2, w * hv.z);
    atomic_add_f32(op + 3, w * hv.w);
}

// ---------------------------------------------------------------------------
extern "C" void kernel_launch(void* const* d_in, const int* in_sizes, int n_in,
                              void* d_out, int out_size, void* d_ws, size_t ws_size,
                              hipStream_t stream) {
    const float*     x  = (const float*)d_in[0];
    const long long* ei = (const long long*)d_in[1];   // [2, E] int64
    const float*     W  = (const float*)d_in[2];
    const float*     b  = (const float*)d_in[3];

    const int D = 128;
    const int N = in_sizes[0] / D;
    const int E = in_sizes[1] / 2;
    const long long* src = ei;
    const long long* dst = ei + E;

    float* out = (float*)d_out;
    float* h   = (float*)d_ws;                     // N*128 floats
    float* deg = (float*)d_ws + (size_t)N * D;     // N floats

    // 1) zero out and deg
    {
        size_t n = (size_t)out_size;
        zero_kernel<<<(unsigned)((n + 255) / 256), 256, 0, stream>>>(out, n);
        zero_kernel<<<(unsigned)((N + 255) / 256), 256, 0, stream>>>(deg, (size_t)N);
    }
    // 2) degrees
    degree_kernel<<<(E + 255) / 256, 256, 0, stream>>>(dst, deg, E);
    // 3) h = x @ W^T + b  (WMMA)
    {
        int strips = (N + 15) / 16;
        int blocks = (strips + 7) / 8;
        gemm_wmma_kernel<<<blocks, 256, 0, stream>>>(x, W, b, h, N);
    }
    // 4) edge scatter with degree normalization
    scatter_kernel<<<(E + 7) / 8, 256, 0, stream>>>(src, dst, h, deg, out, E);
}